// ObjectMatchingModule_90426241450574
// MI455X (gfx1250) — compile-verified
//
#include <hip/hip_runtime.h>
#include <math.h>

#define M_DIM 4096
#define N_DIM 4096
#define D_DIM 256
#define MP1   4097
#define NP1   4097
#define LDK   4097            // K stored unpadded in d_out's K region
#define EPS_INV 10.0f         // 1 / SINKHORN_EPSILON
#define ITERS 100
#define CHUNKS 64
#define ROWS_PER_CHUNK 65     // ceil(4097/64)
#define PLD 4104              // padded stride for partial buffer

typedef __attribute__((ext_vector_type(2))) float v2f;
typedef __attribute__((ext_vector_type(8))) float v8f;

// ---------------------------------------------------------------------------
// GEMM: S = A * B^T using V_WMMA_F32_16X16X4_F32, fused K0 = exp(S/eps).
// One wave computes a 16x64 output strip (4 WMMA tiles, A regs reused 4x).
// A 16x4 f32 layout: lanes 0-15 hold K=0(+1), lanes 16-31 hold K=2(+3).
// ---------------------------------------------------------------------------
__global__ __launch_bounds__(256) void gemm_exp_kernel(const float* __restrict__ A,
                                                       const float* __restrict__ B,
                                                       float* __restrict__ K)
{
    const int lane = threadIdx.x & 31;
    const int wave = threadIdx.x >> 5;
    const int w    = blockIdx.x * 8 + wave;        // 0..16383
    const int m0   = (w & 255) * 16;               // 256 M tiles
    const int n0   = (w >> 8) * 64;                // 64 N strips

    const int row  = lane & 15;
    const int koff = (lane >> 4) * 2;              // 0 or 2

    const float* arow = A + (size_t)(m0 + row) * D_DIM + koff;
    const float* brow = B + (size_t)(n0 + row) * D_DIM + koff;

    v8f acc0 = {}; v8f acc1 = {}; v8f acc2 = {}; v8f acc3 = {};

    for (int k = 0; k < D_DIM; k += 4) {
        v2f a;  a.x  = arow[k];                 a.y  = arow[k + 1];
        v2f b0; b0.x = brow[k];                 b0.y = brow[k + 1];
        v2f b1; b1.x = brow[16 * D_DIM + k];    b1.y = brow[16 * D_DIM + k + 1];
        v2f b2; b2.x = brow[32 * D_DIM + k];    b2.y = brow[32 * D_DIM + k + 1];
        v2f b3; b3.x = brow[48 * D_DIM + k];    b3.y = brow[48 * D_DIM + k + 1];

        acc0 = __builtin_amdgcn_wmma_f32_16x16x4_f32(false, a, false, b0, (short)0, acc0, false, false);
        acc1 = __builtin_amdgcn_wmma_f32_16x16x4_f32(false, a, false, b1, (short)0, acc1, false, false);
        acc2 = __builtin_amdgcn_wmma_f32_16x16x4_f32(false, a, false, b2, (short)0, acc2, false, false);
        acc3 = __builtin_amdgcn_wmma_f32_16x16x4_f32(false, a, false, b3, (short)0, acc3, false, false);
    }

    // C/D layout: VGPR r, lanes 0-15 -> M=r, N=lane; lanes 16-31 -> M=r+8, N=lane-16
    const int mi0 = m0 + ((lane >> 4) << 3);
    const int col = lane & 15;
#pragma unroll
    for (int r = 0; r < 8; ++r) {
        float* out = K + (size_t)(mi0 + r) * LDK + n0 + col;
        out[0]  = expf(acc0[r] * EPS_INV);
        out[16] = expf(acc1[r] * EPS_INV);
        out[32] = expf(acc2[r] * EPS_INV);
        out[48] = expf(acc3[r] * EPS_INV);
    }
}

// Fill augmented last row/col with exp(z/eps); init v = 1.
__global__ void edge_init_kernel(float* __restrict__ K, float* __restrict__ v,
                                 const float* __restrict__ zp)
{
    int i = blockIdx.x * blockDim.x + threadIdx.x;
    float ez = expf(zp[0] * EPS_INV);
    if (i < MP1) {
        K[(size_t)i * LDK + N_DIM] = ez;            // last column (incl. corner)
        if (i < N_DIM) K[(size_t)M_DIM * LDK + i] = ez;  // last row
        v[i] = 1.0f;
    }
}

// u_i = 1 / sum_j K0[i,j] * v[j]   (one block per row, LDS tree reduce)
__global__ __launch_bounds__(256) void rowsum_kernel(const float* __restrict__ K,
                                                     const float* __restrict__ v,
                                                     float* __restrict__ u)
{
    __shared__ float sdata[256];
    const float* kr = K + (size_t)blockIdx.x * LDK;
    float s = 0.0f;
    for (int j = threadIdx.x; j < NP1; j += 256) s += kr[j] * v[j];
    sdata[threadIdx.x] = s;
    __syncthreads();
    for (int off = 128; off > 0; off >>= 1) {
        if ((int)threadIdx.x < off) sdata[threadIdx.x] += sdata[threadIdx.x + off];
        __syncthreads();
    }
    if (threadIdx.x == 0) u[blockIdx.x] = 1.0f / sdata[0];
}

// partial[c, j] = sum_{i in chunk c} K0[i,j] * u[i]   (coalesced in j)
__global__ __launch_bounds__(256) void colsum_partial_kernel(const float* __restrict__ K,
                                                             const float* __restrict__ u,
                                                             float* __restrict__ partial)
{
    int j = blockIdx.x * 256 + threadIdx.x;
    int c = blockIdx.y;
    if (j >= NP1) return;
    int i0 = c * ROWS_PER_CHUNK;
    int i1 = i0 + ROWS_PER_CHUNK; if (i1 > MP1) i1 = MP1;
    float s = 0.0f;
    for (int i = i0; i < i1; ++i) s += K[(size_t)i * LDK + j] * u[i];
    partial[(size_t)c * PLD + j] = s;
}

// v_j = 1 / sum_c partial[c, j]
__global__ __launch_bounds__(256) void colsum_finish_kernel(const float* __restrict__ partial,
                                                            float* __restrict__ v)
{
    int j = blockIdx.x * 256 + threadIdx.x;
    if (j >= NP1) return;
    float s = 0.0f;
#pragma unroll 8
    for (int c = 0; c < CHUNKS; ++c) s += partial[(size_t)c * PLD + j];
    v[j] = 1.0f / s;
}

// K <- diag(u) K0 diag(v) in place; P = K[:M,:N]
__global__ __launch_bounds__(256) void finalize_kernel(float* __restrict__ K,
                                                       float* __restrict__ P,
                                                       const float* __restrict__ u,
                                                       const float* __restrict__ v)
{
    size_t idx = (size_t)blockIdx.x * 256 + threadIdx.x;
    const size_t total = (size_t)MP1 * NP1;
    if (idx >= total) return;
    int i = (int)(idx / NP1);
    int j = (int)(idx - (size_t)i * NP1);
    float val = u[i] * K[idx] * v[j];
    K[idx] = val;
    if (i < M_DIM && j < N_DIM) P[(size_t)i * N_DIM + j] = val;
}

extern "C" void kernel_launch(void* const* d_in, const int* in_sizes, int n_in,
                              void* d_out, int out_size, void* d_ws, size_t ws_size,
                              hipStream_t stream)
{
    (void)in_sizes; (void)n_in; (void)out_size; (void)ws_size;
    const float* A  = (const float*)d_in[0];   // d_M_q [4096,256]
    const float* B  = (const float*)d_in[1];   // d_N_r [4096,256]
    const float* zp = (const float*)d_in[2];   // scalar z

    float* P = (float*)d_out;                          // [4096*4096]
    float* K = P + (size_t)M_DIM * N_DIM;              // [4097*4097], doubles as K0 store

    float* ws      = (float*)d_ws;
    float* u       = ws;                               // 4104
    float* v       = ws + PLD;                         // 4104
    float* partial = ws + 2 * PLD;                     // CHUNKS * PLD

    // 1) K0 = exp(S/eps) via fp32 WMMA GEMM (16384 waves, 8 waves/block)
    gemm_exp_kernel<<<2048, 256, 0, stream>>>(A, B, K);
    // 2) augmented edges + v = 1
    edge_init_kernel<<<(MP1 + 255) / 256, 256, 0, stream>>>(K, v, zp);

    // 3) 100 factorized Sinkhorn iterations: two L2-resident matvecs each
    for (int it = 0; it < ITERS; ++it) {
        rowsum_kernel<<<MP1, 256, 0, stream>>>(K, v, u);
        dim3 g((NP1 + 255) / 256, CHUNKS);
        colsum_partial_kernel<<<g, 256, 0, stream>>>(K, u, partial);
        colsum_finish_kernel<<<(NP1 + 255) / 256, 256, 0, stream>>>(partial, v);
    }

    // 4) materialize K and P
    size_t total = (size_t)MP1 * NP1;
    finalize_kernel<<<(unsigned)((total + 255) / 256), 256, 0, stream>>>(K, P, u, v);
}